// _TopKPoolWrapper_39926015984332
// MI455X (gfx1250) — compile-verified
//
#include <hip/hip_runtime.h>
#include <math.h>

typedef float v2f __attribute__((ext_vector_type(2)));
typedef float v8f __attribute__((ext_vector_type(8)));

#define B_  64
#define N_  2048
#define D_  128
#define K_  1024
#define E_  4194304
#define BN  (B_ * N_)   /* 131072 */
#define BK  (B_ * K_)   /*  65536 */

// d_out layout (float element offsets), outputs concatenated in return order:
// x_out [BK*D], edge_index_out [2*E], batch_out [BK], perm [BK], score_sel [BK], edge_mask [E]
#define O_X     0
#define O_EDGE  (BK * D_)            /*  8388608 */
#define O_BATCH (O_EDGE + 2 * E_)    /* 16777216 */
#define O_PERM  (O_BATCH + BK)
#define O_SCORE (O_PERM + BK)
#define O_MASK  (O_SCORE + BK)

// Monotonic int mapping of f32 (order-preserving), lossless inverse.
__device__ __forceinline__ unsigned int ord32(float f) {
    unsigned int u = __float_as_uint(f);
    return (u & 0x80000000u) ? ~u : (u | 0x80000000u);
}
__device__ __forceinline__ float unord32(unsigned int o) {
    unsigned int u = (o & 0x80000000u) ? (o ^ 0x80000000u) : ~o;
    return __uint_as_float(u);
}

// ---------------------------------------------------------------------------
// K1: score[i] = tanh((x[i]·w)/||w||) via V_WMMA_F32_16X16X4_F32.
//     One wave per 16-row tile of x; B matrix = w chunk replicated over all
//     16 columns, accumulate 32 chunks of K=4 to cover D=128 (all in f32).
//     Also initializes new_idx to -1.
// ---------------------------------------------------------------------------
__global__ void score_kernel(const float* __restrict__ x,
                             const float* __restrict__ w,
                             float* __restrict__ score,
                             int* __restrict__ new_idx) {
    const int tid  = threadIdx.x;
    const int gtid = blockIdx.x * blockDim.x + tid;
    if (gtid < BN) new_idx[gtid] = -1;

    // ||w|| (uniform; w is 512B, stays in cache)
    float ss = 0.0f;
#pragma unroll 8
    for (int k = 0; k < D_; ++k) { float wv = w[k]; ss += wv * wv; }
    const float inv_norm = 1.0f / sqrtf(ss);

    const int wave  = tid >> 5;
    const int lane  = tid & 31;
    const int tile  = blockIdx.x * (blockDim.x >> 5) + wave;  // 8192 tiles total
    const int row   = tile * 16 + (lane & 15);
    const int khalf = (lane >> 4) * 2;                        // 0 or 2

    const float* xr = x + (size_t)row * D_;

    v8f c = {0.f, 0.f, 0.f, 0.f, 0.f, 0.f, 0.f, 0.f};
#pragma unroll 4
    for (int kc = 0; kc < 32; ++kc) {
        const int kb = kc * 4;
        // A (16x4 f32): lane L -> row L&15; VGPR0 = K=khalf, VGPR1 = K=khalf+1
        v2f a;
        a.x = xr[kb + khalf];
        a.y = xr[kb + khalf + 1];
        // B (4x16 f32), all columns identical: VGPR v holds rows v (lanes 0-15)
        // and v+2 (lanes 16-31) -> same khalf indexing as A.
        v2f b;
        b.x = w[kb + khalf];
        b.y = w[kb + khalf + 1];
        c = __builtin_amdgcn_wmma_f32_16x16x4_f32(
                /*neg_a=*/false, a, /*neg_b=*/false, b,
                /*c_mod=*/(short)0, c, /*reuse_a=*/false, /*reuse_b=*/false);
    }

    // Every column of D is identical; lanes 0 and 16 hold rows 0..7 / 8..15.
    if ((lane & 15) == 0) {
        const int base = tile * 16 + (lane >> 4) * 8;
#pragma unroll
        for (int v = 0; v < 8; ++v)
            score[base + v] = tanhf(c[v] * inv_norm);
    }
}

// ---------------------------------------------------------------------------
// K2: per-graph bitonic sort (descending) of 2048 packed keys in LDS.
//     key = ord32(score)<<32 | ~index  -> lax.top_k semantics (ties: lowest
//     index first). Emits perm / score_sel / batch_out, scatters new_idx.
// ---------------------------------------------------------------------------
__global__ void sort_kernel(const float* __restrict__ score,
                            int* __restrict__ perm_i,
                            int* __restrict__ new_idx,
                            float* __restrict__ out) {
    __shared__ unsigned long long keys[N_];
    const int b = blockIdx.x;
    const int t = threadIdx.x;
    const float* sb = score + b * N_;

    for (int i = t; i < N_; i += blockDim.x) {
        unsigned long long o = ord32(sb[i]);
        keys[i] = (o << 32) | (unsigned int)(~(unsigned int)i);
    }
    __syncthreads();

    for (int k = 2; k <= N_; k <<= 1) {
        for (int j = k >> 1; j > 0; j >>= 1) {
            for (int i = t; i < N_; i += blockDim.x) {
                const int ixj = i ^ j;
                if (ixj > i) {
                    unsigned long long a  = keys[i];
                    unsigned long long bb = keys[ixj];
                    const bool desc = ((i & k) == 0);
                    if (desc ? (a < bb) : (a > bb)) {
                        keys[i] = bb;
                        keys[ixj] = a;
                    }
                }
            }
            __syncthreads();
        }
    }

    for (int j = t; j < K_; j += blockDim.x) {
        const unsigned long long key = keys[j];
        const unsigned int idx_local = ~(unsigned int)(key & 0xFFFFFFFFull);
        const float val = unord32((unsigned int)(key >> 32));
        const int pg      = b * N_ + (int)idx_local;   // global node id
        const int out_row = b * K_ + j;                // compacted id
        perm_i[out_row]        = pg;
        new_idx[pg]            = out_row;
        out[O_PERM  + out_row] = (float)pg;
        out[O_SCORE + out_row] = val;
        out[O_BATCH + out_row] = (float)b;
    }
}

// ---------------------------------------------------------------------------
// K3: x_out[r] = x[perm[r]] * score_sel[r]; 32 lanes x float4 per row.
// ---------------------------------------------------------------------------
__global__ void gather_kernel(const float* __restrict__ x,
                              const int* __restrict__ perm_i,
                              const float* __restrict__ score_sel,
                              float* __restrict__ x_out) {
    const int gtid = blockIdx.x * blockDim.x + threadIdx.x;
    const int row  = gtid >> 5;
    const int lane = gtid & 31;
    if (row >= BK) return;
    const int   p = perm_i[row];
    const float s = score_sel[row];
    float4 v = ((const float4*)(x + (size_t)p * D_))[lane];
    v.x *= s; v.y *= s; v.z *= s; v.w *= s;
    ((float4*)(x_out + (size_t)row * D_))[lane] = v;
}

// ---------------------------------------------------------------------------
// K4: relabel edges through new_idx, emit masked edge_index_out + edge_mask.
// ---------------------------------------------------------------------------
__global__ void edge_kernel(const int* __restrict__ edge_index,
                            const int* __restrict__ new_idx,
                            float* __restrict__ out) {
    const int e = blockIdx.x * blockDim.x + threadIdx.x;
    if (e >= E_) return;
    const int nr = new_idx[edge_index[e]];
    const int nc = new_idx[edge_index[E_ + e]];
    const bool m = (nr >= 0) && (nc >= 0);
    out[O_EDGE + e]       = (float)(m ? nr : 0);
    out[O_EDGE + E_ + e]  = (float)(m ? nc : 0);
    out[O_MASK + e]       = m ? 1.0f : 0.0f;
}

extern "C" void kernel_launch(void* const* d_in, const int* in_sizes, int n_in,
                              void* d_out, int out_size, void* d_ws, size_t ws_size,
                              hipStream_t stream) {
    const float* x          = (const float*)d_in[0];
    const int*   edge_index = (const int*)  d_in[1];
    /* d_in[2] = batch: implicit (node i -> i/N), unused */
    const float* w          = (const float*)d_in[3];
    float* out = (float*)d_out;

    char*  ws      = (char*)d_ws;
    float* score   = (float*)ws;                              // BN floats
    int*   new_idx = (int*)(ws + (size_t)BN * sizeof(float)); // BN ints
    int*   perm_i  = (int*)(ws + (size_t)2 * BN * sizeof(float)); // BK ints

    // K1: 8192 tiles of 16 rows, 8 waves/block -> 1024 blocks
    score_kernel<<<BN / 16 / 8, 256, 0, stream>>>(x, w, score, new_idx);
    // K2: one workgroup per graph
    sort_kernel<<<B_, 256, 0, stream>>>(score, perm_i, new_idx, out);
    // K3: BK rows, 32 threads/row
    gather_kernel<<<(BK * 32) / 256, 256, 0, stream>>>(x, perm_i, out + O_SCORE, out + O_X);
    // K4: one thread per edge
    edge_kernel<<<E_ / 256, 256, 0, stream>>>(edge_index, new_idx, out);
}